// ScaledDotProductAttention_75247827026525
// MI455X (gfx1250) — compile-verified
//
#include <hip/hip_runtime.h>

typedef float v2f __attribute__((ext_vector_type(2)));
typedef float v8f __attribute__((ext_vector_type(8)));

// D = A(16x4 f32) * B(4x16 f32) + C(16x16 f32)
#define WMMA_F32X4(a, b, c) \
    __builtin_amdgcn_wmma_f32_16x16x4_f32(false, (a), false, (b), (short)0, (c), false, false)

constexpr int B_ = 4, H_ = 16, S_ = 2048, D_ = 64;
constexpr int WAVES = 8;          // waves per block (wave32)
constexpr int LDS_STRIDE = 20;    // 16 + 4 pad -> bank-conflict-free transpose

__global__ __launch_bounds__(256, 1)
void sdpa_fused_kernel(const float* __restrict__ q,   // [B,H,S,D]
                       const float* __restrict__ kT,  // [B,H,D,S] (pre-transposed)
                       const float* __restrict__ v,   // [B,H,S,D]
                       const int*   __restrict__ mask,// [B,1,S,S]
                       float* __restrict__ out,       // [B,H,S,D]
                       float* __restrict__ attn)      // [B,H,S,S]
{
    __shared__ float plds[WAVES][16 * LDS_STRIDE];

    const int wave = threadIdx.x >> 5;   // wave32
    const int lane = threadIdx.x & 31;
    const int hi   = lane >> 4;          // half-wave: selects K pair / M high half
    const int lm   = lane & 15;

    const int tile = blockIdx.x * WAVES + wave;       // 0..8191
    const int mt   = tile & (S_ / 16 - 1);            // row-tile within S (0..127)
    const int bh   = tile >> 7;                       // fused (b,h) 0..63
    const int b    = bh >> 4;

    const float* qp  = q  + ((size_t)bh * S_ + (size_t)mt * 16) * D_;
    const float* kp  = kT + (size_t)bh * D_ * S_;
    const float* vp  = v  + (size_t)bh * S_ * D_;
    const int*   mp  = mask + (size_t)b * S_ * S_ + (size_t)(mt * 16) * S_;
    float* outp      = out  + ((size_t)bh * S_ + (size_t)mt * 16) * D_;
    float* attnp     = attn + (size_t)bh * S_ * S_ + (size_t)(mt * 16) * S_;

    // ---- Q tile resident in A-layout: chunk c holds K = 4c + 2*hi + {0,1}, row m = lm
    v2f qa[16];
    {
        const float* qrow = qp + (size_t)lm * D_;
#pragma unroll
        for (int c = 0; c < 16; ++c)
            qa[c] = *(const v2f*)(qrow + 4 * c + 2 * hi);
    }

    v8f oacc[4];
#pragma unroll
    for (int t = 0; t < 4; ++t)
        oacc[t] = (v8f){0.f, 0.f, 0.f, 0.f, 0.f, 0.f, 0.f, 0.f};

    float mrun[8], lrun[8];
#pragma unroll
    for (int r = 0; r < 8; ++r) { mrun[r] = -3.402823466e38f; lrun[r] = 0.f; }

    float* myp = &plds[wave][0];

    for (int nt = 0; nt < S_ / 16; ++nt) {
        const int s0 = nt * 16;

        // ---- S-tile = Q @ K^T : 16 chained f32 WMMAs over K=64
        v8f acc = (v8f){0.f, 0.f, 0.f, 0.f, 0.f, 0.f, 0.f, 0.f};
#pragma unroll
        for (int c = 0; c < 16; ++c) {
            const float* kb = kp + (size_t)(4 * c + 2 * hi) * S_ + s0 + lm;
            v2f bv; bv.x = kb[0]; bv.y = kb[S_];   // B-layout: n=lm, k=2*hi+{0,1}
            acc = WMMA_F32X4(qa[c], bv, acc);
        }

        // ---- scale, mask, export attn_score, online-softmax stats (C-layout)
        float corr[8];
#pragma unroll
        for (int r = 0; r < 8; ++r) {
            const int mrow = r + 8 * hi;           // row within 16-row tile
            float sval = acc[r] * 0.125f;          // / sqrt(64)
            const int mk = mp[(size_t)mrow * S_ + s0 + lm];
            sval = (mk == 0) ? -1.0e10f : sval;
            attnp[(size_t)mrow * S_ + s0 + lm] = sval;

            // row max across the 16 lanes holding this row
            float rmax = sval;
            rmax = fmaxf(rmax, __shfl_xor(rmax, 1));
            rmax = fmaxf(rmax, __shfl_xor(rmax, 2));
            rmax = fmaxf(rmax, __shfl_xor(rmax, 4));
            rmax = fmaxf(rmax, __shfl_xor(rmax, 8));

            const float mold = mrun[r];
            const float mnew = fmaxf(mold, rmax);
            const float cr   = __expf(mold - mnew);
            const float p    = __expf(sval - mnew);

            float rsum = p;
            rsum += __shfl_xor(rsum, 1);
            rsum += __shfl_xor(rsum, 2);
            rsum += __shfl_xor(rsum, 4);
            rsum += __shfl_xor(rsum, 8);

            lrun[r] = lrun[r] * cr + rsum;
            mrun[r] = mnew;
            corr[r] = cr;

            // stage P into LDS (C-layout write)
            myp[mrow * LDS_STRIDE + lm] = p;
        }

        // rescale output accumulators per-row by exp(m_old - m_new)
#pragma unroll
        for (int t = 0; t < 4; ++t)
#pragma unroll
            for (int r = 0; r < 8; ++r)
                oacc[t][r] *= corr[r];

        // ---- reload P in A-layout (per-wave LDS slice; same-wave DS ops are in-order)
        v2f pa[4];
#pragma unroll
        for (int c = 0; c < 4; ++c)
            pa[c] = *(const v2f*)(myp + lm * LDS_STRIDE + 4 * c + 2 * hi);

        // ---- O += P @ V : 4 N-tiles x 4 K-chunks of f32 WMMA
#pragma unroll
        for (int t = 0; t < 4; ++t) {
            v8f o = oacc[t];
#pragma unroll
            for (int c = 0; c < 4; ++c) {
                const float* vb = vp + (size_t)(s0 + 4 * c + 2 * hi) * D_ + t * 16 + lm;
                v2f bv; bv.x = vb[0]; bv.y = vb[D_];
                o = WMMA_F32X4(pa[c], bv, o);
            }
            oacc[t] = o;
        }
    }

    // ---- normalize and store output
    float inv[8];
#pragma unroll
    for (int r = 0; r < 8; ++r) inv[r] = 1.0f / lrun[r];
#pragma unroll
    for (int t = 0; t < 4; ++t)
#pragma unroll
        for (int r = 0; r < 8; ++r)
            outp[(size_t)(r + 8 * hi) * D_ + t * 16 + lm] = oacc[t][r] * inv[r];
}

extern "C" void kernel_launch(void* const* d_in, const int* in_sizes, int n_in,
                              void* d_out, int out_size, void* d_ws, size_t ws_size,
                              hipStream_t stream)
{
    const float* q    = (const float*)d_in[0];
    const float* kT   = (const float*)d_in[1];
    const float* v    = (const float*)d_in[2];
    const int*   mask = (const int*)d_in[3];

    float* out  = (float*)d_out;                                   // [B,H,S,D]
    float* attn = out + (size_t)B_ * H_ * S_ * D_;                 // [B,H,S,S]

    const int tiles  = B_ * H_ * (S_ / 16);                        // 8192
    const int blocks = tiles / WAVES;                              // 1024

    sdpa_fused_kernel<<<blocks, WAVES * 32, 0, stream>>>(q, kT, v, mask, out, attn);
}